// FaAnchor_Loss_76991583748730
// MI455X (gfx1250) — compile-verified
//
#include <hip/hip_runtime.h>

typedef __attribute__((ext_vector_type(2))) float v2f;
typedef __attribute__((ext_vector_type(8))) float v8f;

#define N_SAMPLES 1000000
#define DIM       128
#define NCLASS    1000

// Workspace layout (floats): ws[0..999] = per-class distance sums,
//                            ws[1000..1999] = per-class counts.

__global__ void fa_init_ws(float* __restrict__ ws) {
    int i = blockIdx.x * blockDim.x + threadIdx.x;
    if (i < 2 * NCLASS) ws[i] = 0.0f;
}

// One wave processes 16 consecutive samples per batch.
// A-matrix (16x4 f32, per ISA layout): lane l holds M = l&15,
// K = k0 + 2*(l>>4) and K+1  -> a contiguous float2 per lane.
// B = all-ones, so C[m][n] = sum_k sq[m][k] accumulates per-sample
// squared distances in f32 across the 32 K-chunks of D=128.
__global__ __launch_bounds__(256) void fa_anchor_dist_kernel(
    const float* __restrict__ x,
    const float* __restrict__ anchors,
    const long long* __restrict__ y,
    float* __restrict__ ws)
{
    __shared__ float lsum[NCLASS];
    __shared__ float lcnt[NCLASS];
    for (int c = threadIdx.x; c < NCLASS; c += blockDim.x) {
        lsum[c] = 0.0f;
        lcnt[c] = 0.0f;
    }
    __syncthreads();

    const int lane          = threadIdx.x & 31;
    const int waveInBlock   = threadIdx.x >> 5;
    const int wavesPerBlock = blockDim.x >> 5;
    const int globalWave    = blockIdx.x * wavesPerBlock + waveInBlock;
    const int totalWaves    = gridDim.x * wavesPerBlock;

    const int m    = lane & 15;        // sample-in-batch this lane feeds
    const int koff = (lane >> 4) << 1; // 0 for lanes 0-15, 2 for lanes 16-31

    const int nBatches = N_SAMPLES / 16; // 62500, exact

    const v2f onesB = {1.0f, 1.0f};

    for (int batch = globalWave; batch < nBatches; batch += totalWaves) {
        const int    i    = batch * 16 + m;
        const int    cls  = (int)y[i];
        const float* xrow = x + (size_t)i * DIM + koff;
        const float* arow = anchors + (size_t)cls * DIM + koff;

        v8f acc = {};
        #pragma unroll 8
        for (int it = 0; it < DIM / 4; ++it) {
            v2f xv = *(const v2f*)(xrow + it * 4);
            v2f av = *(const v2f*)(arow + it * 4);
            v2f d  = xv - av;
            v2f s  = d * d;          // elementwise square in VALU (packed f32)
            // C += A(16x4 sq-diff) x B(4x16 ones): row-sum reduction in f32
            acc = __builtin_amdgcn_wmma_f32_16x16x4_f32(
                false, s, false, onesB, (short)0, acc, false, false);
        }

        // All N columns of acc are identical. Per C/D layout:
        // lanes 0-15 hold d_0..d_7 in acc[0..7]; lanes 16-31 hold d_8..d_15.
        const int   r    = lane & 7;
        const float dval = acc[r];
        if ((lane & 8) == 0) {                        // dedupe duplicate columns
            const int myS   = (lane < 16) ? r : (8 + r);
            const int myCls = (int)y[batch * 16 + myS];
            atomicAdd(&lsum[myCls], dval);
            atomicAdd(&lcnt[myCls], 1.0f);
        }
    }

    __syncthreads();
    for (int c = threadIdx.x; c < NCLASS; c += blockDim.x) {
        float s = lsum[c];
        float n = lcnt[c];
        if (n != 0.0f) {
            atomicAdd(&ws[c], s);
            atomicAdd(&ws[NCLASS + c], n);
        }
    }
}

__global__ __launch_bounds__(256) void fa_finalize_kernel(
    const float* __restrict__ ws, float* __restrict__ out)
{
    __shared__ float red[256];
    float v = 0.0f;
    for (int c = threadIdx.x; c < NCLASS; c += 256) {
        float s = ws[c];
        float n = ws[NCLASS + c];
        if (n > 0.0f) v += s / (n * n);
    }
    red[threadIdx.x] = v;
    __syncthreads();
    for (int off = 128; off > 0; off >>= 1) {
        if ((int)threadIdx.x < off) red[threadIdx.x] += red[threadIdx.x + off];
        __syncthreads();
    }
    if (threadIdx.x == 0) out[0] = red[0];
}

extern "C" void kernel_launch(void* const* d_in, const int* in_sizes, int n_in,
                              void* d_out, int out_size, void* d_ws, size_t ws_size,
                              hipStream_t stream) {
    const float*     x       = (const float*)d_in[0];
    const float*     anchors = (const float*)d_in[1];
    const long long* y       = (const long long*)d_in[2]; // int64 labels
    float*           out     = (float*)d_out;
    float*           ws      = (float*)d_ws;

    fa_init_ws<<<(2 * NCLASS + 255) / 256, 256, 0, stream>>>(ws);
    // 2048 blocks x 8 waves = 16384 waves; ~3.8 batches (16 samples) per wave,
    // consecutive batches to consecutive waves for sequential HBM streaming.
    fa_anchor_dist_kernel<<<2048, 256, 0, stream>>>(x, anchors, y, ws);
    fa_finalize_kernel<<<1, 256, 0, stream>>>(ws, out);
}